// GATLayer_652835029725
// MI455X (gfx1250) — compile-verified
//
#include <hip/hip_runtime.h>

// GAT layer, MI455X (gfx1250, wave32).
// Reference simplifies exactly: softmax over an axis of length 1 == 1.0, so
//   out = segment_sum(z[col], row)  with  z = x @ W_fc^T.
// Phase 1: fp32 WMMA GEMM (V_WMMA_F32_16X16X4_F32), W_fc staged in LDS.
// Phase 2: per-edge gather of z + native fp32 atomics into out.
// d_ws holds z: 100000*64*4 = 25.6 MB (assumed <= ws_size).

#define N_NODES 100000
#define N_EDGES 1600000
#define IN_DIM  128
#define OUT_DIM 64
#define M_TILES (N_NODES / 16)   // 6250, exact

typedef __attribute__((ext_vector_type(2))) float v2f;
typedef __attribute__((ext_vector_type(8))) float v8f;

// ---------------------------------------------------------------- zero out
__global__ void gat_zero_kernel(float4* __restrict__ out) {
  int i = blockIdx.x * blockDim.x + threadIdx.x;
  if (i < N_NODES * OUT_DIM / 4) {
    float4 zf = {0.f, 0.f, 0.f, 0.f};
    out[i] = zf;
  }
}

// ------------------------------------------------- z = x @ W^T via WMMA f32
// One wave -> one 16-row tile of z, all 64 output columns (4 accumulators).
// A 16x4 f32 frag: lane (m = lane&15, kh = 2*(lane>>4)) holds x[row][kk+kh..+1].
// B 4x16 f32 frag: same lane holds W[n=m][kk+kh..+1] (B[k][n] = W_fc[n][k]).
__global__ void __launch_bounds__(256)
gat_gemm_wmma_kernel(const float* __restrict__ x, const float* __restrict__ W,
                     float* __restrict__ z) {
  __shared__ float sW[OUT_DIM * IN_DIM];          // 32 KB of 320 KB LDS
  {
    const float4* src = (const float4*)W;
    float4* dst = (float4*)sW;
    for (int i = threadIdx.x; i < OUT_DIM * IN_DIM / 4; i += 256)
      dst[i] = src[i];
  }
  __syncthreads();

  const int lane = threadIdx.x & 31;
  const int wave = threadIdx.x >> 5;
  const int tile = blockIdx.x * 8 + wave;
  if (tile >= M_TILES) return;

  const int m  = lane & 15;              // row within tile (A), col within tile (B)
  const int kh = (lane >> 4) << 1;       // K sub-offset: 0 or 2

  const float* xrow = x + (size_t)(tile * 16 + m) * IN_DIM + kh;
  const float* w0 = sW + ( 0 + m) * IN_DIM + kh;
  const float* w1 = sW + (16 + m) * IN_DIM + kh;
  const float* w2 = sW + (32 + m) * IN_DIM + kh;
  const float* w3 = sW + (48 + m) * IN_DIM + kh;

  v8f acc0 = {}, acc1 = {}, acc2 = {}, acc3 = {};

  #pragma unroll 8
  for (int kk = 0; kk < IN_DIM; kk += 4) {
    v2f a  = *(const v2f*)(xrow + kk);   // global_load_b64
    v2f b0 = *(const v2f*)(w0 + kk);     // ds_load_b64
    v2f b1 = *(const v2f*)(w1 + kk);
    v2f b2 = *(const v2f*)(w2 + kk);
    v2f b3 = *(const v2f*)(w3 + kk);
    acc0 = __builtin_amdgcn_wmma_f32_16x16x4_f32(false, a, false, b0, (short)0, acc0, false, false);
    acc1 = __builtin_amdgcn_wmma_f32_16x16x4_f32(false, a, false, b1, (short)0, acc1, false, false);
    acc2 = __builtin_amdgcn_wmma_f32_16x16x4_f32(false, a, false, b2, (short)0, acc2, false, false);
    acc3 = __builtin_amdgcn_wmma_f32_16x16x4_f32(false, a, false, b3, (short)0, acc3, false, false);
  }

  // C/D layout: VGPR v -> row v (lanes 0-15) / row v+8 (lanes 16-31), col = m.
  const int rbase = tile * 16 + (lane >> 4) * 8;
  float* zp = z + (size_t)rbase * OUT_DIM + m;
  #pragma unroll
  for (int v = 0; v < 8; ++v) {
    zp[(size_t)v * OUT_DIM +  0] = acc0[v];
    zp[(size_t)v * OUT_DIM + 16] = acc1[v];
    zp[(size_t)v * OUT_DIM + 32] = acc2[v];
    zp[(size_t)v * OUT_DIM + 48] = acc3[v];
  }
}

// ------------------------------------------- out[row[e]] += z[col[e]] (alpha==1)
// One thread handles 2 consecutive floats of one edge; z gather hits L2
// (z is 25.6 MB << 192 MB), atomics are native global_atomic_add_f32.
__global__ void gat_scatter_kernel(const int* __restrict__ ei,
                                   const float* __restrict__ z,
                                   float* __restrict__ out) {
  long long i = (long long)blockIdx.x * blockDim.x + threadIdx.x;
  if (i >= (long long)N_EDGES * (OUT_DIM / 2)) return;
  const int e  = (int)(i >> 5);
  const int n2 = (int)(i & 31) * 2;
  const int r = ei[e];                 // edge_index[0][e]
  const int c = ei[N_EDGES + e];       // edge_index[1][e]
  if ((unsigned)r >= N_NODES || (unsigned)c >= N_NODES) return;  // safety net
  v2f zv = *(const v2f*)(z + (size_t)c * OUT_DIM + n2);
  unsafeAtomicAdd(out + (size_t)r * OUT_DIM + n2,     zv.x);
  unsafeAtomicAdd(out + (size_t)r * OUT_DIM + n2 + 1, zv.y);
}

// ---------------------------------------------------------------- launcher
extern "C" void kernel_launch(void* const* d_in, const int* in_sizes, int n_in,
                              void* d_out, int out_size, void* d_ws, size_t ws_size,
                              hipStream_t stream) {
  const float* x    = (const float*)d_in[0];   // (100000, 128) f32
  const int*   ei   = (const int*)  d_in[1];   // (2, 1600000)
  const float* W_fc = (const float*)d_in[2];   // (64, 128) f32
  (void)d_in[3];                               // W_attn: dead (softmax over len-1 axis == 1)
  (void)in_sizes; (void)n_in; (void)out_size; (void)ws_size;

  float* out = (float*)d_out;                  // (100000, 64) f32
  float* z   = (float*)d_ws;                   // scratch: 25.6 MB

  // 1) zero the accumulation target (harness poisons d_out; scatter accumulates)
  gat_zero_kernel<<<(N_NODES * OUT_DIM / 4 + 255) / 256, 256, 0, stream>>>((float4*)out);

  // 2) z = x @ W_fc^T  (6250 16-row tiles, 8 waves/block)
  gat_gemm_wmma_kernel<<<(M_TILES + 7) / 8, 256, 0, stream>>>(x, W_fc, z);

  // 3) out[row] += z[col]
  const long long items = (long long)N_EDGES * (OUT_DIM / 2);
  gat_scatter_kernel<<<(unsigned)((items + 255) / 256), 256, 0, stream>>>(ei, z, out);
}